// SIREConvBase_56118042689984
// MI455X (gfx1250) — compile-verified
//
#include <hip/hip_runtime.h>
#include <hip/hip_bf16.h>

// ---------------------------------------------------------------------------
// SIREConvBase GNN conv for gfx1250 (MI455X), fp32 end-to-end.
//   out[v] = sum_{e: dst[e]=v} norm[src]*norm[dst] * ( [nf[dst]||nf[src]] + ef[e] ) @ W + b scaled
// Edge GEMM (16 edges x 128 x 64) runs on V_WMMA_F32_16X16X4_F32.
// ---------------------------------------------------------------------------

typedef __attribute__((ext_vector_type(2))) float v2f;
typedef __attribute__((ext_vector_type(8))) float v8f;

#define TILE_E 16          // edges per WMMA tile (M dimension)
#define KDIM   128         // 2*D
#define NDIM   64          // D
#define WAVES  4           // one wave per 16-wide N tile
#define BLOCK  (32 * WAVES)
#define TILES_PER_BLOCK 8
#define MSG_STRIDE 132     // 128 + 4 pad floats: conflict-free ds_load_b64 A reads

// ---------------- helper kernels -------------------------------------------

__global__ void zero_f32_kernel(float* __restrict__ p, int n) {
    int i = blockIdx.x * blockDim.x + threadIdx.x;
    if (i < n) p[i] = 0.0f;
}

__global__ void degree_kernel(const int* __restrict__ edge_dst, float* __restrict__ deg, int n_edges) {
    int e = blockIdx.x * blockDim.x + threadIdx.x;
    if (e < n_edges) atomicAdd(&deg[edge_dst[e]], 1.0f);
}

__global__ void norm_kernel(float* __restrict__ degnorm, int n_nodes) {
    int i = blockIdx.x * blockDim.x + threadIdx.x;
    if (i < n_nodes) degnorm[i] = 1.0f / sqrtf(fmaxf(degnorm[i], 1.0f));
}

// ---------------- main edge-GEMM + scatter kernel --------------------------

__global__ __launch_bounds__(BLOCK)
void edge_gemm_scatter_kernel(const float* __restrict__ nfeat,   // [N,64]
                              const float* __restrict__ efeat,   // [E,128]
                              const int*   __restrict__ esrc,
                              const int*   __restrict__ edst,
                              const float* __restrict__ W,       // [128,64]
                              const float* __restrict__ bias,    // [64]
                              const float* __restrict__ norm,    // [N]
                              float*       __restrict__ out,     // [N,64]
                              int n_tiles) {
    __shared__ float smsg[TILE_E * MSG_STRIDE];  // 16 x 128 staged message (padded)
    __shared__ float sscale[TILE_E];             // norm[src]*norm[dst] per edge
    __shared__ int   sdst[TILE_E];

    const int tid  = threadIdx.x;
    const int wave = tid >> 5;      // 0..3 -> N tile
    const int lane = tid & 31;
    const int l16  = lane & 15;
    const int hi   = lane >> 4;     // 0/1 -> K sub-pair / M upper half

    // --- Preload this wave's B fragments of W (held in 64 VGPRs). ---
    // 16x16x4 B layout (4xK rows, 16 N cols): lane l -> N = l&15,
    // K = 4*kt + 2*hi + {0,1}.
    const int ncol = wave * 16 + l16;
    v2f Bfrag[32];
#pragma unroll
    for (int kt = 0; kt < 32; ++kt) {
        int k0 = kt * 4 + hi * 2;
        Bfrag[kt].x = W[(k0    ) * NDIM + ncol];
        Bfrag[kt].y = W[(k0 + 1) * NDIM + ncol];
    }
    const float bn = bias[ncol];

    const int tile0 = blockIdx.x * TILES_PER_BLOCK;
    for (int t = 0; t < TILES_PER_BLOCK; ++t) {
        const int tile = tile0 + t;          // uniform across block -> EXEC stays all-1
        if (tile >= n_tiles) break;
        const int ebase = tile * TILE_E;

        // --- Stage msg[16][128] = [nf[dst] || nf[src]] + ef into LDS. ---
        // Thread tid: edge m = tid>>3, 16-float chunk c = tid&7 (float4 x4).
        {
            const int m = tid >> 3;
            const int c = tid & 7;
            const int e = ebase + m;
            const int s = esrc[e];
            const int d = edst[e];
            if (tid < TILE_E) {
                int ee = ebase + tid;
                sscale[tid] = norm[esrc[ee]] * norm[edst[ee]];
                sdst[tid]   = edst[ee];
            }
            const float4* ef = (const float4*)(efeat + (size_t)e * KDIM + c * 16);
            const float4* nf = (c < 4)
                ? (const float4*)(nfeat + (size_t)d * NDIM + c * 16)
                : (const float4*)(nfeat + (size_t)s * NDIM + (c - 4) * 16);
            float* dst = &smsg[m * MSG_STRIDE + c * 16];
#pragma unroll
            for (int i = 0; i < 4; ++i) {
                float4 ev = ef[i];
                float4 nv = nf[i];
                float4 r;
                r.x = ev.x + nv.x; r.y = ev.y + nv.y;
                r.z = ev.z + nv.z; r.w = ev.w + nv.w;
                *(float4*)(dst + i * 4) = r;
            }
        }
        __syncthreads();

        // --- C[16x16] += A[16x128] * W[:, 16w:16w+16] via 32 fp32 WMMAs. ---
        // A layout: lane l -> M = l&15, VGPR0 = K=4kt+2hi, VGPR1 = K+1.
        v8f C = {};
#pragma unroll
        for (int kt = 0; kt < 32; ++kt) {
            const int k0 = kt * 4 + hi * 2;
            v2f A = *(const v2f*)&smsg[l16 * MSG_STRIDE + k0];
            C = __builtin_amdgcn_wmma_f32_16x16x4_f32(
                    false, A, false, Bfrag[kt], (short)0, C, false, false);
        }

        // --- Bias, norm scale, scatter-add. C: vgpr r, lane l -> M=r+8*hi, N=l&15. ---
#pragma unroll
        for (int r = 0; r < 8; ++r) {
            const int m = r + hi * 8;
            const float val = (C[r] + bn) * sscale[m];
            atomicAdd(&out[(size_t)sdst[m] * NDIM + ncol], val);
        }
        __syncthreads();
    }
}

// Scalar fallback for a non-multiple-of-16 edge tail (none for E=1.6M).
__global__ void edge_tail_kernel(const float* __restrict__ nfeat, const float* __restrict__ efeat,
                                 const int* __restrict__ esrc, const int* __restrict__ edst,
                                 const float* __restrict__ W, const float* __restrict__ bias,
                                 const float* __restrict__ norm, float* __restrict__ out,
                                 int e0, int n_edges) {
    int idx = blockIdx.x * blockDim.x + threadIdx.x;
    int e = e0 + (idx >> 6);
    int n = idx & 63;
    if (e >= n_edges) return;
    int s = esrc[e], d = edst[e];
    float acc = bias[n];
    for (int k = 0; k < NDIM; ++k) {
        float a0 = nfeat[(size_t)d * NDIM + k] + efeat[(size_t)e * KDIM + k];
        float a1 = nfeat[(size_t)s * NDIM + k] + efeat[(size_t)e * KDIM + NDIM + k];
        acc += a0 * W[k * NDIM + n] + a1 * W[(k + NDIM) * NDIM + n];
    }
    atomicAdd(&out[(size_t)d * NDIM + n], acc * norm[s] * norm[d]);
}

// ---------------- launch ----------------------------------------------------

extern "C" void kernel_launch(void* const* d_in, const int* in_sizes, int n_in,
                              void* d_out, int out_size, void* d_ws, size_t ws_size,
                              hipStream_t stream) {
    const float* nfeat = (const float*)d_in[0];
    const float* efeat = (const float*)d_in[1];
    const int*   esrc  = (const int*)  d_in[2];
    const int*   edst  = (const int*)  d_in[3];
    const float* W     = (const float*)d_in[4];
    const float* bias  = (const float*)d_in[5];
    float* out = (float*)d_out;

    const int n_nodes = in_sizes[0] / NDIM;   // HEADS*D = 64 floats per node
    const int n_edges = in_sizes[2];

    float* degnorm = (float*)d_ws;            // n_nodes floats: degree -> norm

    // Zero accumulators every call (graph replays re-accumulate).
    zero_f32_kernel<<<(out_size + 255) / 256, 256, 0, stream>>>(out, out_size);
    zero_f32_kernel<<<(n_nodes + 255) / 256, 256, 0, stream>>>(degnorm, n_nodes);

    degree_kernel<<<(n_edges + 255) / 256, 256, 0, stream>>>(edst, degnorm, n_edges);
    norm_kernel<<<(n_nodes + 255) / 256, 256, 0, stream>>>(degnorm, n_nodes);

    const int n_tiles = n_edges / TILE_E;
    if (n_tiles > 0) {
        const int grid = (n_tiles + TILES_PER_BLOCK - 1) / TILES_PER_BLOCK;
        edge_gemm_scatter_kernel<<<grid, BLOCK, 0, stream>>>(
            nfeat, efeat, esrc, edst, W, bias, degnorm, out, n_tiles);
    }
    const int tail = n_edges - n_tiles * TILE_E;
    if (tail > 0) {
        edge_tail_kernel<<<(tail * NDIM + 255) / 256, 256, 0, stream>>>(
            nfeat, efeat, esrc, edst, W, bias, degnorm, out, n_tiles * TILE_E, n_edges);
    }
}